// ChildSumTreeLSTMCell_84722524881532
// MI455X (gfx1250) — compile-verified
//
#include <hip/hip_runtime.h>

// ---------------------------------------------------------------------------
// ChildSum TreeLSTM for MI455X (gfx1250, wave32).
// All GEMMs run on V_WMMA_F32_16X16X4_F32 (fp32 in/out, matches reference
// dtype). Workload is HBM-bound (~2 GB traffic vs ~41 GFLOP), so fp32 WMMA
// is the right precision/perf point.
// ---------------------------------------------------------------------------

typedef __attribute__((ext_vector_type(2))) float v2f;
typedef __attribute__((ext_vector_type(8))) float v8f;

#define LDS_PITCH 66   // 64-col chunk + 2 pad floats -> conflict-free column reads

__device__ __forceinline__ v8f wmma_f32_k4(v2f a, v2f b, v8f c) {
  // D = A(16x4) * B(4x16) + C(16x16), fp32
  return __builtin_amdgcn_wmma_f32_16x16x4_f32(false, a, false, b, (short)0, c,
                                               false, false);
}

__device__ __forceinline__ float sigmoidf(float x) {
  return 1.0f / (1.0f + __expf(-x));
}

__device__ __forceinline__ void atomic_add_f32(float* p, float v) {
  (void)__hip_atomic_fetch_add(p, v, __ATOMIC_RELAXED, __HIP_MEMORY_SCOPE_AGENT);
}

__device__ __forceinline__ v8f splat8(float b) {
  v8f v;
#pragma unroll
  for (int j = 0; j < 8; ++j) v[j] = b;
  return v;
}

// ---------------------------------------------------------------------------
// Kernel 0: zero h_sum (workspace) and d_out (c_tilde accumulator half).
// ---------------------------------------------------------------------------
__global__ __launch_bounds__(256) void zero_kernel(float* __restrict__ hsum,
                                                   float* __restrict__ out,
                                                   long n) {
  long i = (long)blockIdx.x * 256 + threadIdx.x;
  if (i < n) {
    hsum[i] = 0.0f;
    out[i] = 0.0f;
  }
}

// ---------------------------------------------------------------------------
// Kernel 1: per-edge message.
//   edge_w = sf*df + W_eoh[:,t] + b_eoh            [E,64]
//   msg    = edge_w @ W_el.T + b_el                [E,128]  (WMMA)
//   h_sum[dst, 0:128] += h_src * msg               (fp32 atomics)
// One wave = 16 edges. W_el (128x64) staged in LDS, pitch 66.
// ---------------------------------------------------------------------------
__global__ __launch_bounds__(256) void edge_msg_kernel(
    const float* __restrict__ h_src, const float* __restrict__ sfeat,
    const float* __restrict__ dfeat, const float* __restrict__ W_eoh,
    const float* __restrict__ b_eoh, const float* __restrict__ W_el,
    const float* __restrict__ b_el, const int* __restrict__ etype,
    const int* __restrict__ dst_idx, float* __restrict__ hsum, int E) {
  __shared__ float lds[128 * LDS_PITCH];
  // stage W_el [128 x 64]
  for (int idx = threadIdx.x; idx < 128 * 64; idx += 256) {
    int h = idx >> 6, g = idx & 63;
    lds[h * LDS_PITCH + g] = W_el[h * 64 + g];
  }
  __syncthreads();

  int lane = threadIdx.x & 31;
  int wave = threadIdx.x >> 5;
  int tile = blockIdx.x * 8 + wave;
  int numTiles = (E + 15) >> 4;
  if (tile >= numTiles) return;  // no barriers after this point

  int eBase = tile << 4;
  int m = lane & 15;          // A row
  int half = lane >> 4;       // which K pair
  int koff = half << 1;       // K offset within a 4-step
  int e = min(eBase + m, E - 1);
  int t = etype[e];

  // Build fp32 A fragments for K = 0..63 (16 WMMA steps).
  v2f a[16];
  const float* sf = sfeat + (long)e * 64;
  const float* df = dfeat + (long)e * 64;
#pragma unroll
  for (int k = 0; k < 16; ++k) {
    int g = (k << 2) + koff;
    v2f av;
    av.x = sf[g] * df[g] + W_eoh[g * 3 + t] + b_eoh[g];
    av.y = sf[g + 1] * df[g + 1] + W_eoh[(g + 1) * 3 + t] + b_eoh[g + 1];
    a[k] = av;
  }

  int n = lane & 15;  // C/D column within tile
#pragma unroll
  for (int nt = 0; nt < 8; ++nt) {
    int h = (nt << 4) + n;
    v8f c = splat8(b_el[h]);
    const float* wrow = &lds[h * LDS_PITCH];
#pragma unroll
    for (int k = 0; k < 16; ++k) {
      v2f b = *(const v2f*)(wrow + (k << 2) + koff);
      c = wmma_f32_k4(a[k], b, c);
    }
    // scale by h_src and scatter-add into h_sum[:, 0:128]
#pragma unroll
    for (int j = 0; j < 8; ++j) {
      int e2 = eBase + j + (half << 3);
      if (e2 < E) {
        float val = c[j] * h_src[(long)e2 * 128 + h];
        int d = dst_idx[e2];
        atomic_add_f32(&hsum[(long)d * 256 + h], val);
      }
    }
  }
}

// ---------------------------------------------------------------------------
// Kernel 2: h_sum[dst, 128:256] += embed_dst   (coalesced fp32 atomics)
// ---------------------------------------------------------------------------
__global__ __launch_bounds__(256) void edge_embed_kernel(
    const float* __restrict__ embed, const int* __restrict__ dst_idx,
    float* __restrict__ hsum, long total) {
  long i = (long)blockIdx.x * 256 + threadIdx.x;
  if (i < total) {
    int e = (int)(i >> 7);
    int x = (int)(i & 127);
    int d = dst_idx[e];
    atomic_add_f32(&hsum[(long)d * 256 + 128 + x], embed[i]);
  }
}

// ---------------------------------------------------------------------------
// Shared gate helper: acc(16x128) += h_sum_tile(16x256) @ W.T(256x128) via
// fp32 WMMA. W staged through LDS in four 64-wide K chunks; all waves of the
// block execute the barriers uniformly (tail waves compute with clamped rows).
// ---------------------------------------------------------------------------
__device__ __forceinline__ void gate16(const float* __restrict__ hsum,
                                       const float* __restrict__ W,
                                       float* lds, int rowA, int n, int koff,
                                       v8f acc[8]) {
  for (int kc = 0; kc < 4; ++kc) {
    __syncthreads();  // previous chunk / previous gate fully consumed
    for (int idx = threadIdx.x; idx < 128 * 64; idx += 256) {
      int h = idx >> 6, g = idx & 63;
      lds[h * LDS_PITCH + g] = W[h * 256 + (kc << 6) + g];
    }
    __syncthreads();
    const float* arow = hsum + (long)rowA * 256 + (kc << 6) + koff;
#pragma unroll
    for (int k = 0; k < 16; ++k) {
      v2f a2 = *(const v2f*)(arow + (k << 2));
#pragma unroll
      for (int nt = 0; nt < 8; ++nt) {
        v2f b2 = *(const v2f*)(lds + ((nt << 4) + n) * LDS_PITCH + (k << 2) + koff);
        acc[nt] = wmma_f32_k4(a2, b2, acc[nt]);
      }
    }
  }
  __syncthreads();
}

// ---------------------------------------------------------------------------
// Kernel 3: f = sigmoid(h_sum @ Wf.T + bWf + bf) -> stashed in d_out[:, 0:128]
// ---------------------------------------------------------------------------
__global__ __launch_bounds__(256) void node_f_kernel(
    const float* __restrict__ hsum, const float* __restrict__ Wf,
    const float* __restrict__ bWf, const float* __restrict__ bf,
    float* __restrict__ out, int N) {
  __shared__ float lds[128 * LDS_PITCH];
  int lane = threadIdx.x & 31;
  int wave = threadIdx.x >> 5;
  int tile = blockIdx.x * 8 + wave;
  int numTiles = (N + 15) >> 4;
  bool active = tile < numTiles;
  int nodeBase = (active ? tile : 0) << 4;
  int n = lane & 15;
  int half = lane >> 4;
  int koff = half << 1;
  int rowA = min(nodeBase + n, N - 1);

  v8f acc[8];
#pragma unroll
  for (int nt = 0; nt < 8; ++nt) {
    int h = (nt << 4) + n;
    acc[nt] = splat8(bWf[h] + bf[h]);
  }
  gate16(hsum, Wf, lds, rowA, n, koff, acc);

#pragma unroll
  for (int nt = 0; nt < 8; ++nt) {
    int h = (nt << 4) + n;
#pragma unroll
    for (int j = 0; j < 8; ++j) {
      int node = nodeBase + j + (half << 3);
      if (active && node < N) out[(long)node * 256 + h] = sigmoidf(acc[nt][j]);
    }
  }
}

// ---------------------------------------------------------------------------
// Kernel 4: c_tilde accumulation: d_out[dst, 128:256] += f[dst] * c_src
// ---------------------------------------------------------------------------
__global__ __launch_bounds__(256) void edge_ctilde_kernel(
    const float* __restrict__ c_src, const int* __restrict__ dst_idx,
    float* __restrict__ out, long total) {
  long i = (long)blockIdx.x * 256 + threadIdx.x;
  if (i < total) {
    int e = (int)(i >> 7);
    int h = (int)(i & 127);
    int d = dst_idx[e];
    float fv = out[(long)d * 256 + h];  // f stashed in [:, 0:128]
    atomic_add_f32(&out[(long)d * 256 + 128 + h], fv * c_src[i]);
  }
}

// ---------------------------------------------------------------------------
// Kernel 5: i,u,o gates (WMMA) + final combine:
//   c = i*u + c_tilde ; h = o * tanh(c) ; d_out = [h | c]
// ---------------------------------------------------------------------------
__global__ __launch_bounds__(256) void node_final_kernel(
    const float* __restrict__ hsum, const float* __restrict__ Wi,
    const float* __restrict__ bWi, const float* __restrict__ bi,
    const float* __restrict__ Wu, const float* __restrict__ bWu,
    const float* __restrict__ bu, const float* __restrict__ Wo,
    const float* __restrict__ bWo, const float* __restrict__ bo,
    float* __restrict__ out, int N) {
  __shared__ float lds[128 * LDS_PITCH];
  int lane = threadIdx.x & 31;
  int wave = threadIdx.x >> 5;
  int tile = blockIdx.x * 8 + wave;
  int numTiles = (N + 15) >> 4;
  bool active = tile < numTiles;
  int nodeBase = (active ? tile : 0) << 4;
  int n = lane & 15;
  int half = lane >> 4;
  int koff = half << 1;
  int rowA = min(nodeBase + n, N - 1);

  // i gate -> sigmoid
  v8f gi[8];
#pragma unroll
  for (int nt = 0; nt < 8; ++nt) {
    int h = (nt << 4) + n;
    gi[nt] = splat8(bWi[h] + bi[h]);
  }
  gate16(hsum, Wi, lds, rowA, n, koff, gi);
#pragma unroll
  for (int nt = 0; nt < 8; ++nt)
#pragma unroll
    for (int j = 0; j < 8; ++j) gi[nt][j] = sigmoidf(gi[nt][j]);

  // u gate -> tanh, fold into gi = i*u
  {
    v8f gu[8];
#pragma unroll
    for (int nt = 0; nt < 8; ++nt) {
      int h = (nt << 4) + n;
      gu[nt] = splat8(bWu[h] + bu[h]);
    }
    gate16(hsum, Wu, lds, rowA, n, koff, gu);
#pragma unroll
    for (int nt = 0; nt < 8; ++nt)
#pragma unroll
      for (int j = 0; j < 8; ++j) gi[nt][j] *= tanhf(gu[nt][j]);
  }

  // o gate -> sigmoid
  v8f go[8];
#pragma unroll
  for (int nt = 0; nt < 8; ++nt) {
    int h = (nt << 4) + n;
    go[nt] = splat8(bWo[h] + bo[h]);
  }
  gate16(hsum, Wo, lds, rowA, n, koff, go);

  // combine + store [h | c]
#pragma unroll
  for (int nt = 0; nt < 8; ++nt) {
    int h = (nt << 4) + n;
#pragma unroll
    for (int j = 0; j < 8; ++j) {
      int node = nodeBase + j + (half << 3);
      if (active && node < N) {
        long base = (long)node * 256;
        float c = gi[nt][j] + out[base + 128 + h];  // + c_tilde
        float hv = sigmoidf(go[nt][j]) * tanhf(c);
        out[base + h] = hv;
        out[base + 128 + h] = c;
      }
    }
  }
}

// ---------------------------------------------------------------------------
// Host launch
// ---------------------------------------------------------------------------
extern "C" void kernel_launch(void* const* d_in, const int* in_sizes, int n_in,
                              void* d_out, int out_size, void* d_ws,
                              size_t ws_size, hipStream_t stream) {
  const float* h_src = (const float*)d_in[0];
  const float* c_src = (const float*)d_in[1];
  const float* embed = (const float*)d_in[2];
  const float* sfeat = (const float*)d_in[3];
  const float* dfeat = (const float*)d_in[4];
  const float* Wf = (const float*)d_in[5];
  const float* bWf = (const float*)d_in[6];
  const float* bf = (const float*)d_in[7];
  const float* Wi = (const float*)d_in[8];
  const float* bWi = (const float*)d_in[9];
  const float* bi = (const float*)d_in[10];
  const float* Wu = (const float*)d_in[11];
  const float* bWu = (const float*)d_in[12];
  const float* bu = (const float*)d_in[13];
  const float* Wo = (const float*)d_in[14];
  const float* bWo = (const float*)d_in[15];
  const float* bo = (const float*)d_in[16];
  const float* W_eoh = (const float*)d_in[17];
  const float* b_eoh = (const float*)d_in[18];
  const float* W_el = (const float*)d_in[19];
  const float* b_el = (const float*)d_in[20];
  const int* etype = (const int*)d_in[21];
  const int* dst_idx = (const int*)d_in[22];

  int E = in_sizes[0] / 128;  // h_src is [E,128]
  int N = out_size / 256;     // out is [N,256]
  float* hsum = (float*)d_ws; // [N,256] fp32
  float* out = (float*)d_out;

  long zn = (long)N * 256;
  zero_kernel<<<(unsigned)((zn + 255) / 256), 256, 0, stream>>>(hsum, out, zn);

  int etiles = (E + 15) >> 4;
  edge_msg_kernel<<<(etiles + 7) / 8, 256, 0, stream>>>(
      h_src, sfeat, dfeat, W_eoh, b_eoh, W_el, b_el, etype, dst_idx, hsum, E);

  long en = (long)E * 128;
  edge_embed_kernel<<<(unsigned)((en + 255) / 256), 256, 0, stream>>>(
      embed, dst_idx, hsum, en);

  int ntiles = (N + 15) >> 4;
  node_f_kernel<<<(ntiles + 7) / 8, 256, 0, stream>>>(hsum, Wf, bWf, bf, out, N);

  edge_ctilde_kernel<<<(unsigned)((en + 255) / 256), 256, 0, stream>>>(
      c_src, dst_idx, out, en);

  node_final_kernel<<<(ntiles + 7) / 8, 256, 0, stream>>>(
      hsum, Wi, bWi, bi, Wu, bWu, bu, Wo, bWo, bo, out, N);
}